// MACE_63977832841988
// MI455X (gfx1250) — compile-verified
//
#include <hip/hip_runtime.h>

typedef __attribute__((ext_vector_type(2))) float v2f;
typedef __attribute__((ext_vector_type(8))) float v8f;

#define MUL   128
#define DIN   9
#define KTOT  219   // 9 + 45 + 165 symmetric monomials
#define KPAD  220   // multiple of 4 (K-step = 4 for V_WMMA_F32_16X16X4_F32)
#define MONS  222   // monomial row stride in LDS (even -> 8B aligned pairs)
#define QCOL  32    // padded output columns (24 used)

// ---------------------------------------------------------------------------
// Prep: symmetrize the bases into M[k, q]  (KPAD x QCOL, zero padded).
// k: monomial index (deg1: 0..8, deg2: 9..53, deg3: 54..218, 219 = pad)
// q: 0=nu1/out0, 1..3=nu1/out1(p), 4..5=nu2/out0(s), 6..11=nu2/out1(s,p),
//    12..14=nu3/out0(s), 15..23=nu3/out1(s,p), 24..31 = pad
// Deterministic: each thread owns exactly one M entry.
// ---------------------------------------------------------------------------
__global__ void MACE_build_M(const float* __restrict__ b10,
                             const float* __restrict__ b11,
                             const float* __restrict__ b20,
                             const float* __restrict__ b21,
                             const float* __restrict__ b30,
                             const float* __restrict__ b31,
                             float* __restrict__ Mws) {
  int idx = blockIdx.x * blockDim.x + threadIdx.x;
  if (idx >= KPAD * QCOL) return;
  int k = idx / QCOL;
  int q = idx % QCOL;
  float v = 0.0f;

  if (k < 9) {                                   // degree-1 monomial x_a
    int a = k;
    if (q == 0)            v = b10[a];           // (9,1,1)
    else if (q <= 3)       v = b11[a * 3 + (q - 1)]; // (9,1,3)
  } else if (k < 54) {                           // degree-2 monomial x_A x_B, A<=B
    int want = k - 9;
    int A = 0, B = 0, cnt = 0;
    for (int a = 0; a < 9; ++a)
      for (int b = a; b < 9; ++b) { if (cnt == want) { A = a; B = b; } ++cnt; }
    if (q >= 4 && q < 12) {
      int s, p, pd;
      const float* basis;
      if (q < 6) { s = q - 4; p = 0; basis = b20; pd = 1; }          // (9,9,2,1)
      else       { s = (q - 6) / 3; p = (q - 6) % 3; basis = b21; pd = 3; } // (9,9,2,3)
      for (int i = 0; i < 9; ++i)
        for (int j = 0; j < 9; ++j) {
          int lo = i < j ? i : j, hi = i < j ? j : i;
          if (lo == A && hi == B)
            v += basis[((i * 9 + j) * 2 + s) * pd + p];
        }
    }
  } else if (k < 219) {                          // degree-3 monomial, A<=B<=C
    int want = k - 54;
    int A = 0, B = 0, C = 0, cnt = 0;
    for (int a = 0; a < 9; ++a)
      for (int b = a; b < 9; ++b)
        for (int c = b; c < 9; ++c) { if (cnt == want) { A = a; B = b; C = c; } ++cnt; }
    if (q >= 12 && q < 24) {
      int s, p, pd;
      const float* basis;
      if (q < 15) { s = q - 12; p = 0; basis = b30; pd = 1; }          // (9,9,9,3,1)
      else        { s = (q - 15) / 3; p = (q - 15) % 3; basis = b31; pd = 3; } // (9,9,9,3,3)
      for (int i = 0; i < 9; ++i)
        for (int j = 0; j < 9; ++j)
          for (int l = 0; l < 9; ++l) {
            int s0 = i, s1 = j, s2 = l, t;
            if (s0 > s1) { t = s0; s0 = s1; s1 = t; }
            if (s1 > s2) { t = s1; s1 = s2; s2 = t; }
            if (s0 > s1) { t = s0; s0 = s1; s1 = t; }
            if (s0 == A && s1 == B && s2 == C)
              v += basis[(((i * 9 + j) * 9 + l) * 3 + s) * pd + p];
          }
    }
  }
  Mws[idx] = v;   // rows k==219 and cols q>=24 stay zero
}

// ---------------------------------------------------------------------------
// Main kernel: one block per node, 256 threads = 8 waves (wave32).
// Wave w owns channel rows [16w, 16w+16): builds P = mon @ M with
// V_WMMA_F32_16X16X4_F32 (2 N-tiles x 55 K-steps), then per-row epilogue
// contracts P with per-species weights.
// ---------------------------------------------------------------------------
__global__ __launch_bounds__(256) void MACE_symcon_kernel(
    const float* __restrict__ nf,      // (2048, 1152)
    const int*   __restrict__ species, // (2048,)
    const float* __restrict__ Mws,     // (KPAD, QCOL)
    const float* __restrict__ w10, const float* __restrict__ w11,
    const float* __restrict__ w20, const float* __restrict__ w21,
    const float* __restrict__ w30, const float* __restrict__ w31,
    float* __restrict__ out) {         // (2048, 512)
  __shared__ float monA[MUL * MONS];        // 113.7 KB: monomials per channel
  __shared__ float Mp[(KPAD / 2) * 64];     //  28.2 KB: M in (k-pair, n) layout
  __shared__ float Pb[MUL * 33];            //  16.9 KB: GEMM result P[r, q]

  const int t    = threadIdx.x;
  const int node = blockIdx.x;
  const float* nfi = nf + node * (MUL * DIN);

  // ---- Stage M into LDS, interleaving k-pairs so B operand = one b64 read.
  for (int idx = t; idx < KPAD * QCOL; idx += 256) {
    int k = idx >> 5, n = idx & 31;
    Mp[(k >> 1) * 64 + n * 2 + (k & 1)] = Mws[idx];
  }

  // ---- Build symmetric monomials for this node's 128 channels.
  if (t < MUL) {
    const int m = t;
    float x[9];
    x[0] = nfi[m];                                        // l=0 block
#pragma unroll
    for (int a = 0; a < 3; ++a) x[1 + a] = nfi[128 + m * 3 + a];  // l=1 block
#pragma unroll
    for (int a = 0; a < 5; ++a) x[4 + a] = nfi[512 + m * 5 + a];  // l=2 block

    float* row = &monA[m * MONS];
    int k = 0;
#pragma unroll
    for (int a = 0; a < 9; ++a) row[k++] = x[a];
    float xx[45];
    int kp = 0;
#pragma unroll
    for (int a = 0; a < 9; ++a)
#pragma unroll
      for (int b = a; b < 9; ++b) { xx[kp] = x[a] * x[b]; row[k++] = xx[kp]; ++kp; }
    kp = 0;
#pragma unroll
    for (int a = 0; a < 9; ++a)
#pragma unroll
      for (int b = a; b < 9; ++b) {
        float ab = xx[kp++];
#pragma unroll
        for (int c = b; c < 9; ++c) row[k++] = ab * x[c];
      }
    row[219] = 0.0f; row[220] = 0.0f; row[221] = 0.0f;    // K padding
  }
  __syncthreads();

  // ---- WMMA GEMM: P[16 rows x 32 cols] per wave. EXEC all-ones here.
  const int wave = t >> 5;
  const int lane = t & 31;
  const int rowA = (wave << 4) + (lane & 15);   // A: lane -> M row, halves -> K pair
  const int koff = (lane >> 4) << 1;            // 0 or 2
  const int ncol = (lane & 15);                 // B: lane -> N col, halves -> K pair

  v8f acc0 = {};
  v8f acc1 = {};
#pragma unroll 2
  for (int ks = 0; ks < KPAD / 4; ++ks) {
    const int K0 = ks * 4 + koff;               // even
    v2f av  = *(const v2f*)&monA[rowA * MONS + K0];
    const float* bb = &Mp[(K0 >> 1) * 64];
    v2f bv0 = *(const v2f*)&bb[ncol * 2];
    v2f bv1 = *(const v2f*)&bb[32 + ncol * 2];
    acc0 = __builtin_amdgcn_wmma_f32_16x16x4_f32(false, av, false, bv0,
                                                 (short)0, acc0, false, false);
    acc1 = __builtin_amdgcn_wmma_f32_16x16x4_f32(false, av, false, bv1,
                                                 (short)0, acc1, false, false);
  }

  // ---- Spill C tiles: lane<16 holds rows 0..7, lane>=16 rows 8..15.
  {
    const int rbase = (wave << 4) + ((lane >> 4) << 3);
    const int col   = lane & 15;
#pragma unroll
    for (int v = 0; v < 8; ++v) {
      Pb[(rbase + v) * 33 + col]      = acc0[v];
      Pb[(rbase + v) * 33 + 16 + col] = acc1[v];
    }
  }
  __syncthreads();

  // ---- Epilogue: contract P with per-species weights, scatter to output.
  if (t < MUL) {
    const int m = t;
    const float* P = &Pb[m * 33];
    const int e = species[node];
    float o0 = w10[e * 128 + m] * P[0]
             + w20[(e * 2 + 0) * 128 + m] * P[4]
             + w20[(e * 2 + 1) * 128 + m] * P[5]
             + w30[(e * 3 + 0) * 128 + m] * P[12]
             + w30[(e * 3 + 1) * 128 + m] * P[13]
             + w30[(e * 3 + 2) * 128 + m] * P[14];
    out[node * 512 + m] = o0;
#pragma unroll
    for (int p = 0; p < 3; ++p) {
      float op = w11[e * 128 + m] * P[1 + p]
               + w21[(e * 2 + 0) * 128 + m] * P[6 + p]
               + w21[(e * 2 + 1) * 128 + m] * P[9 + p]
               + w31[(e * 3 + 0) * 128 + m] * P[15 + p]
               + w31[(e * 3 + 1) * 128 + m] * P[18 + p]
               + w31[(e * 3 + 2) * 128 + m] * P[21 + p];
      out[node * 512 + 128 + m * 3 + p] = op;
    }
  }
}

// ---------------------------------------------------------------------------
// d_in order (setup_inputs dict insertion order — b/w interleaved!):
// 0 node_feats, 1 species, 2 b10, 3 w10, 4 b11, 5 w11, 6 b20, 7 w20,
// 8 b21, 9 w21, 10 b30, 11 w30, 12 b31, 13 w31
// ---------------------------------------------------------------------------
extern "C" void kernel_launch(void* const* d_in, const int* in_sizes, int n_in,
                              void* d_out, int out_size, void* d_ws, size_t ws_size,
                              hipStream_t stream) {
  const float* nf   = (const float*)d_in[0];
  const int*   spec = (const int*)d_in[1];
  const float* b10  = (const float*)d_in[2];
  const float* w10  = (const float*)d_in[3];
  const float* b11  = (const float*)d_in[4];
  const float* w11  = (const float*)d_in[5];
  const float* b20  = (const float*)d_in[6];
  const float* w20  = (const float*)d_in[7];
  const float* b21  = (const float*)d_in[8];
  const float* w21  = (const float*)d_in[9];
  const float* b30  = (const float*)d_in[10];
  const float* w30  = (const float*)d_in[11];
  const float* b31  = (const float*)d_in[12];
  const float* w31  = (const float*)d_in[13];
  float* Mws = (float*)d_ws;                 // KPAD*QCOL*4 = 28,160 bytes
  float* out = (float*)d_out;

  MACE_build_M<<<(KPAD * QCOL + 255) / 256, 256, 0, stream>>>(
      b10, b11, b20, b21, b30, b31, Mws);
  MACE_symcon_kernel<<<2048, 256, 0, stream>>>(
      nf, spec, Mws, w10, w11, w20, w21, w30, w31, out);
}